// CrossCorrelationLayer_38139309589253
// MI455X (gfx1250) — compile-verified
//
#include <hip/hip_runtime.h>

// CrossCorrelationLayer for MI455X (gfx1250):
//   out[b,k,t] = sum_c x1[b,c,t] * x2[b,c,t+D-k],  D=10, k in [0,21)
// Implemented as the 21-wide diagonal band of x1^T * x2 per batch using
// V_WMMA_F32_16X16X4_F32 (native fp32 matrix path -> reference numerics).
//
// Roofline: 2.82 GFLOP vs ~537MB minimum traffic -> memory bound at 23.3TB/s
// (~23us). Each wave owns 64 t's (4 row tiles), needs 6 col tiles of x2,
// accumulates 12 16x16 f32 tiles over K=C=512 (128 WMMA K-steps of 4).

typedef __attribute__((ext_vector_type(2))) float v2f;
typedef __attribute__((ext_vector_type(8))) float v8f;

#define B_  16
#define C_  512
#define T_  8192
#define D_  10
#define ND_ 21
#define RT  4          // row tiles (of 16 t) per wave
#define CT  (RT + 2)   // col tiles of x2 needed per wave

__global__ __launch_bounds__(128)
void corr_band_wmma_f32(const float* __restrict__ x1,
                        const float* __restrict__ x2,
                        float* __restrict__ out) {
    const int lane  = threadIdx.x & 31;
    const int wave  = blockIdx.x * 4 + (threadIdx.x >> 5);
    const int spb   = T_ / (16 * RT);        // strips per batch = 128
    const int b     = wave / spb;
    const int t0    = (wave % spb) * (16 * RT);

    const int half = lane >> 4;              // K half: lanes 0-15 K={0,1}, 16-31 K={2,3}
    const int lm   = lane & 15;              // M (A rows) / N (B cols) index

    const size_t xbase = (size_t)b * C_ * T_;
    // Bake the per-half-wave K offset (2*half rows of the channel dim) into the base.
    const float* A0 = x1 + xbase + (size_t)(2 * half) * T_;
    const float* B0 = x2 + xbase + (size_t)(2 * half) * T_;

    // Loop-invariant per-lane coordinates.
    int tA[RT];
#pragma unroll
    for (int r = 0; r < RT; ++r) tA[r] = t0 + 16 * r + lm;

    int   tB[CT];
    float bmask[CT];
#pragma unroll
    for (int j = 0; j < CT; ++j) {
        int col  = t0 - 16 + 16 * j + lm;    // x2 column (t') for this lane
        bmask[j] = (col >= 0 && col < T_) ? 1.0f : 0.0f;
        int cc   = col < 0 ? 0 : col;
        tB[j]    = cc > (T_ - 1) ? (T_ - 1) : cc;  // clamped address, zeroed by mask
    }

    v8f acc[RT][3];
#pragma unroll
    for (int r = 0; r < RT; ++r)
#pragma unroll
        for (int j = 0; j < 3; ++j) acc[r][j] = v8f{};

    // K loop over channels, 4 per WMMA.
    for (int c = 0; c < C_; c += 4) {
        const float* Ac = A0 + (size_t)c * T_;
        const float* Bc = B0 + (size_t)c * T_;

        v2f a[RT];
#pragma unroll
        for (int r = 0; r < RT; ++r) {       // A frag: A[M=lm][K=2*half + {0,1}]
            a[r].x = Ac[tA[r]];
            a[r].y = Ac[tA[r] + T_];
        }

        v2f bf[CT];
#pragma unroll
        for (int j = 0; j < CT; ++j) {       // B frag: B[K=2*half + {0,1}][N=lm]
            bf[j].x = Bc[tB[j]]      * bmask[j];
            bf[j].y = Bc[tB[j] + T_] * bmask[j];
        }

#pragma unroll
        for (int r = 0; r < RT; ++r)
#pragma unroll
            for (int j = 0; j < 3; ++j)
                acc[r][j] = __builtin_amdgcn_wmma_f32_16x16x4_f32(
                    false, a[r], false, bf[r + j],
                    (short)0, acc[r][j], false, false);
    }

    // Scatter the band back out. C/D layout: VGPR e, lane -> M = e + 8*half,
    // N = lm. k = t - t' + D; every needed (k,t) lands in exactly one tile.
    float* outb = out + (size_t)b * ND_ * T_;
#pragma unroll
    for (int r = 0; r < RT; ++r) {
#pragma unroll
        for (int j = 0; j < 3; ++j) {
            const int tc0 = t0 - 16 + 16 * (r + j);
#pragma unroll
            for (int e = 0; e < 8; ++e) {
                int t  = t0 + 16 * r + e + 8 * half;
                int k  = t - (tc0 + lm) + D_;
                if ((unsigned)k < (unsigned)ND_)
                    outb[(size_t)k * T_ + t] = acc[r][j][e];
            }
        }
    }
}

extern "C" void kernel_launch(void* const* d_in, const int* in_sizes, int n_in,
                              void* d_out, int out_size, void* d_ws, size_t ws_size,
                              hipStream_t stream) {
    const float* x1 = (const float*)d_in[0];
    const float* x2 = (const float*)d_in[1];
    float* out = (float*)d_out;

    const int waves  = B_ * (T_ / (16 * RT));  // 2048
    const int blocks = waves / 4;              // 4 waves (128 threads) per block
    corr_band_wmma_f32<<<blocks, dim3(128), 0, stream>>>(x1, x2, out);
}